// ScaledDotProduct_7868380086538
// MI455X (gfx1250) — compile-verified
//
#include <hip/hip_runtime.h>

// ScaledDotProduct (attention fwd, returns context + full attention matrix)
// B=32, N=1024, D=64, fp32.  CDNA5 / gfx1250, wave32, V_WMMA_F32_16X16X4_F32.

typedef float v2f __attribute__((ext_vector_type(2)));
typedef float v4f __attribute__((ext_vector_type(4)));
typedef float v8f __attribute__((ext_vector_type(8)));

#define BB 32
#define NN 1024
#define DD 64
#define SROW 1028   // LDS row stride in floats (1024 + 4 pad vs 64 banks)

__global__ __launch_bounds__(32)
void sdpa_f32_wmma_kernel(const float* __restrict__ q,
                          const float* __restrict__ k,
                          const float* __restrict__ v,
                          const float* __restrict__ scale_p,
                          float* __restrict__ ctx_out,   // [B,N,D]
                          float* __restrict__ attn_out)  // [B,N,N]
{
    // Raw scores / probabilities for this wave's 16 query rows: 16 x 1024 f32.
    __shared__ float sS[16 * SROW];

    const int lane = threadIdx.x;     // 0..31 (wave32)
    const int half = lane >> 4;       // 0 | 1
    const int l15  = lane & 15;

    const int blk   = blockIdx.x;     // b*64 + qtile
    const int b     = blk >> 6;
    const int qtile = blk & 63;
    const int q0    = qtile * 16;

    const float scale = scale_p[0];

    const float* __restrict__ qb = q + (size_t)b * NN * DD;
    const float* __restrict__ kb = k + (size_t)b * NN * DD;
    const float* __restrict__ vb = v + (size_t)b * NN * DD;

    // ---------------- Q tile -> A-operand registers ----------------
    // A (16x4 f32) layout: lanes 0-15: row=l15, v0=K0,v1=K1 ; lanes 16-31: v0=K2,v1=K3.
    // qa[s] covers K = 4*s .. 4*s+3 of the D=64 reduction.
    v2f qa[16];
    {
        const float* __restrict__ qr = qb + (size_t)(q0 + l15) * DD + half * 2;
        #pragma unroll
        for (int s = 0; s < 16; ++s) {
            qa[s][0] = qr[4 * s + 0];
            qa[s][1] = qr[4 * s + 1];
        }
    }

    // ---------------- Pass 1: S = scale * (Q @ K^T), spill to LDS ----------------
    for (int j = 0; j < 64; ++j) {
        // B operand = K^T tile (4 x 16): lane indexes key n=l15; element (kk,n)=K[n][kk]
        // -> identical load pattern to the A operand, over the key rows.
        v2f kt[16];
        {
            const float* __restrict__ kr = kb + (size_t)(j * 16 + l15) * DD + half * 2;
            #pragma unroll
            for (int s = 0; s < 16; ++s) {
                kt[s][0] = kr[4 * s + 0];
                kt[s][1] = kr[4 * s + 1];
            }
        }
        v8f acc = {0.f, 0.f, 0.f, 0.f, 0.f, 0.f, 0.f, 0.f};
        #pragma unroll
        for (int s = 0; s < 16; ++s) {
            acc = __builtin_amdgcn_wmma_f32_16x16x4_f32(
                false, qa[s], false, kt[s], (short)0, acc, false, false);
        }
        // C/D layout: VGPR r -> row = half*8 + r, col = lane-local l15.
        const int col = j * 16 + l15;
        #pragma unroll
        for (int r = 0; r < 8; ++r) {
            sS[(half * 8 + r) * SROW + col] = acc[r] * scale;
        }
    }
    __syncthreads();

    // ---------------- Softmax over each row of 1024 ----------------
    // lane -> row = l15, column half = half*512.
    const int srow  = l15;
    const int cbase = half * 512;
    float* __restrict__ sr = &sS[srow * SROW + cbase];

    float m = -__builtin_inff();
    for (int c = 0; c < 512; c += 4) {
        const v4f s4 = *(const v4f*)&sr[c];
        m = fmaxf(m, fmaxf(fmaxf(s4[0], s4[1]), fmaxf(s4[2], s4[3])));
    }
    m = fmaxf(m, __shfl_xor(m, 16, 32));   // combine the two column halves

    float sum = 0.f;
    for (int c = 0; c < 512; c += 4) {
        v4f s4 = *(const v4f*)&sr[c];
        v4f e4;
        e4[0] = __expf(s4[0] - m);
        e4[1] = __expf(s4[1] - m);
        e4[2] = __expf(s4[2] - m);
        e4[3] = __expf(s4[3] - m);
        *(v4f*)&sr[c] = e4;
        sum += (e4[0] + e4[1]) + (e4[2] + e4[3]);
    }
    sum += __shfl_xor(sum, 16, 32);
    const float inv = 1.f / sum;

    // Normalize in LDS and stream the attention matrix to global.
    float* __restrict__ arow = attn_out + ((size_t)b * NN + q0 + srow) * NN + cbase;
    for (int c = 0; c < 512; c += 4) {
        v4f e4 = *(const v4f*)&sr[c];
        e4[0] *= inv; e4[1] *= inv; e4[2] *= inv; e4[3] *= inv;
        *(v4f*)&sr[c]  = e4;
        *(v4f*)&arow[c] = e4;
    }
    __syncthreads();

    // ---------------- Context: C = P @ V ----------------
    v8f cacc[4];
    #pragma unroll
    for (int dt = 0; dt < 4; ++dt)
        cacc[dt] = (v8f){0.f, 0.f, 0.f, 0.f, 0.f, 0.f, 0.f, 0.f};

    for (int n = 0; n < NN; n += 4) {
        // A operand (P tile 16x4) straight out of LDS in A layout.
        const float* __restrict__ ap = &sS[l15 * SROW + n + half * 2];
        v2f a;
        a[0] = ap[0];
        a[1] = ap[1];
        const int kk = n + half * 2;
        const float* __restrict__ vr0 = vb + (size_t)kk * DD;        // key kk
        const float* __restrict__ vr1 = vr0 + DD;                    // key kk+1
        #pragma unroll
        for (int dt = 0; dt < 4; ++dt) {
            const int d = dt * 16 + l15;
            v2f bv;                    // B operand (V tile 4x16): lane = output dim d
            bv[0] = vr0[d];
            bv[1] = vr1[d];
            cacc[dt] = __builtin_amdgcn_wmma_f32_16x16x4_f32(
                false, a, false, bv, (short)0, cacc[dt], false, false);
        }
    }

    // Store context tile (C/D layout -> [B,N,D]).
    #pragma unroll
    for (int dt = 0; dt < 4; ++dt) {
        const int d = dt * 16 + l15;
        #pragma unroll
        for (int r = 0; r < 8; ++r) {
            const int row = half * 8 + r;
            ctx_out[((size_t)b * NN + q0 + row) * DD + d] = cacc[dt][r];
        }
    }
}

extern "C" void kernel_launch(void* const* d_in, const int* in_sizes, int n_in,
                              void* d_out, int out_size, void* d_ws, size_t ws_size,
                              hipStream_t stream) {
    (void)in_sizes; (void)n_in; (void)d_ws; (void)ws_size; (void)out_size;
    const float* q     = (const float*)d_in[0];
    const float* k     = (const float*)d_in[1];
    const float* v     = (const float*)d_in[2];
    const float* scale = (const float*)d_in[3];

    float* ctx  = (float*)d_out;                          // [B,N,D] first
    float* attn = (float*)d_out + (size_t)BB * NN * DD;   // then [B,N,N]

    dim3 grid(BB * (NN / 16));   // 2048 single-wave workgroups
    dim3 block(32);
    sdpa_f32_wmma_kernel<<<grid, block, 0, stream>>>(q, k, v, scale, ctx, attn);
}